// LinearCRF_30769145708745
// MI455X (gfx1250) — compile-verified
//
#include <hip/hip_runtime.h>

// ---------------------------------------------------------------------------
// Linear-chain CRF forward scores on gfx1250 (MI455X).
//
// unlabeled: forward algorithm done in the *linear* domain with per-row
// running log-scale, so each step is a real (16x32)@(32x32) fp32 GEMM that
// maps onto V_WMMA_F32_16X16X4_F32 (2 N... actually 2 M-tiles x 8 K-tiles =
// 16 WMMA per step). beta is carried as the WMMA B operand; the transition
// matrix exp(T) is stored label-permuted so the WMMA C output registers are
// directly the next step's B operand registers (no LDS / no shuffles on the
// recurrence path).
//
// Label permutation: slot(kt, h, v) holds label
//      lab = 16*(kt>>2) + 8*h + 2*(kt&3) + v        (bijection on 0..31)
// B operand (4x16 f32, K-tile kt):  VGPR v, lane-half h -> K = 2h+v
// C output  (16x16 f32, M-tile jt): VGPR r, lane-half h -> row r+8h (+16jt)
// With A[jt][kt] = exp(T)[lab(kt,h,a)][ (lane&15) + 16*jt ], the output slot
// C[jt=kt>>2][r=2*(kt&3)+v] carries exactly label lab(kt,h,v) -> feeds B[kt].
// ---------------------------------------------------------------------------

#define S_LEN 512
#define L_LAB 32
#define B_SZ  64
#define START_I 29
#define END_I   30

typedef float v2f __attribute__((ext_vector_type(2)));
typedef float v8f __attribute__((ext_vector_type(8)));

__global__ __launch_bounds__(32) void crf_scan_kernel(
    const float* __restrict__ lstm,   // (64,512,32)
    const int*   __restrict__ lens,   // (64,)
    const float* __restrict__ T,      // (32,32)
    float*       __restrict__ ws_unlab) // (64,) per-batch logsumexp result
{
  const int lane = threadIdx.x;        // wave32: one wave per block
  const int m    = lane & 15;          // batch row within tile (N dim of WMMA)
  const int h    = lane >> 4;          // lane half
  const int b    = blockIdx.x * 16 + m;
  const int len_m = lens[b];

  // ---- constant per-wave matrices in registers -------------------------
  // A tiles: A[jt][kt], element a in {0,1}: exp(T[lab(kt,h,a)][m+16*jt])
  v2f A[2][8];
  float eend[2][8];                    // exp(T[q][END]) at C-layout slot q
#pragma unroll
  for (int jt = 0; jt < 2; ++jt) {
#pragma unroll
    for (int kt = 0; kt < 8; ++kt) {
      int lab0 = 16 * (kt >> 2) + 8 * h + 2 * (kt & 3);
      A[jt][kt].x = __expf(T[(lab0 + 0) * L_LAB + (m + 16 * jt)]);
      A[jt][kt].y = __expf(T[(lab0 + 1) * L_LAB + (m + 16 * jt)]);
    }
#pragma unroll
    for (int r = 0; r < 8; ++r)
      eend[jt][r] = __expf(T[(r + 8 * h + 16 * jt) * L_LAB + END_I]);
  }

  // ---- beta init: one-hot at START (permuted B layout) -----------------
  // Synthesizes step t=0: one-hot @ E * exp(lstm[0]) == exp(alpha0). Exact.
  v2f Bm[8];
#pragma unroll
  for (int kt = 0; kt < 8; ++kt) {
    int lab0 = 16 * (kt >> 2) + 8 * h + 2 * (kt & 3);
    Bm[kt].x = (lab0 + 0 == START_I) ? 1.0f : 0.0f;
    Bm[kt].y = (lab0 + 1 == START_I) ? 1.0f : 0.0f;
  }

  const float* rowp = lstm + (size_t)b * (S_LEN * L_LAB);

  // software-pipelined raw lstm loads (exp applied at point of use)
  float raw[16], rawn[16];
#pragma unroll
  for (int jt = 0; jt < 2; ++jt)
#pragma unroll
    for (int r = 0; r < 8; ++r)
      raw[jt * 8 + r] = rowp[(r + 8 * h + 16 * jt)];   // t = 0

  double logscale = 0.0;

  for (int t = 0; t < S_LEN; ++t) {
    // prefetch raw lstm row for t+1 (clamped; last iter redundant/harmless)
    const int tn = (t + 1 < S_LEN) ? (t + 1) : t;
#pragma unroll
    for (int jt = 0; jt < 2; ++jt)
#pragma unroll
      for (int r = 0; r < 8; ++r)
        rawn[jt * 8 + r] = rowp[tn * L_LAB + (r + 8 * h + 16 * jt)];

    // exp(lstm[t]) — TRANS ops, co-execute with the WMMA chain
    float el[16];
#pragma unroll
    for (int i = 0; i < 16; ++i) el[i] = __expf(raw[i]);

    // newT = Eperm x beta : 2 independent accumulation chains of 8 WMMAs
    v8f C0 = {}; v8f C1 = {};
#pragma unroll
    for (int kt = 0; kt < 8; ++kt) {
      C0 = __builtin_amdgcn_wmma_f32_16x16x4_f32(
              false, A[0][kt], false, Bm[kt], (short)0, C0, false, false);
      C1 = __builtin_amdgcn_wmma_f32_16x16x4_f32(
              false, A[1][kt], false, Bm[kt], (short)0, C1, false, false);
    }

    // elementwise emission multiply (still unnormalized alpha_t in linear dom)
    float vv[16];
#pragma unroll
    for (int r = 0; r < 8; ++r) {
      vv[r]     = C0[r] * el[r];
      vv[8 + r] = C1[r] * el[8 + r];
    }

    // row sum across the 32 labels: 16 local slots + the other lane-half
    float rs_half = 0.0f;
#pragma unroll
    for (int i = 0; i < 16; ++i) rs_half += vv[i];
    const float rowsum = rs_half + __shfl_xor(rs_half, 16, 32);

    // capture unlabeled_b at t == len_b - 1 (rare, wave-uniform branch)
    const bool pred = (t + 1 == len_m);
    if (__ballot(pred) != 0ULL) {
      float w_half = 0.0f;
#pragma unroll
      for (int r = 0; r < 8; ++r)
        w_half += vv[r] * eend[0][r] + vv[8 + r] * eend[1][r];
      const float w = w_half + __shfl_xor(w_half, 16, 32);
      if (pred && h == 0)
        ws_unlab[b] = (float)(logscale + (double)__logf(w));
    }

    // renormalize row to sum 1; fold scale into running log
    const float inv = __builtin_amdgcn_rcpf(rowsum);
    logscale += (double)__logf(rowsum);

    // rebuild B operand for next step: pure register remap + scale
#pragma unroll
    for (int kt = 0; kt < 8; ++kt) {
      const int jt = kt >> 2;
      const int r  = 2 * (kt & 3);
      Bm[kt].x = vv[jt * 8 + r]     * inv;
      Bm[kt].y = vv[jt * 8 + r + 1] * inv;
    }

#pragma unroll
    for (int i = 0; i < 16; ++i) raw[i] = rawn[i];
  }
}

// ---------------------------------------------------------------------------
// labeled path: pure gathers, one block per batch element
// ---------------------------------------------------------------------------
__global__ __launch_bounds__(256) void crf_labeled_kernel(
    const float* __restrict__ lstm,
    const int*   __restrict__ lens,
    const int*   __restrict__ tags,
    const float* __restrict__ T,
    float*       __restrict__ ws_lab)
{
  const int b   = blockIdx.x;
  const int tid = threadIdx.x;
  const int len = lens[b];
  const int*   tg = tags + b * S_LEN;
  const float* ls = lstm + (size_t)b * S_LEN * L_LAB;

  float acc = 0.0f;
  for (int t = 1 + tid; t < S_LEN; t += 256) {
    if (t < len) {
      const int cur = tg[t], prev = tg[t - 1];
      acc += T[prev * L_LAB + cur] + ls[t * L_LAB + cur];
    }
  }
  if (tid == 0) {
    const int t0 = tg[0];
    acc += T[START_I * L_LAB + t0] + ls[t0];        // begin
    acc += T[tg[len - 1] * L_LAB + END_I];          // end transition
  }

  __shared__ float red[256];
  red[tid] = acc;
  __syncthreads();
#pragma unroll
  for (int s = 128; s > 0; s >>= 1) {
    if (tid < s) red[tid] += red[tid + s];
    __syncthreads();
  }
  if (tid == 0) ws_lab[b] = red[0];
}

// ---------------------------------------------------------------------------
// final deterministic reduction -> d_out[0]=unlabeled, d_out[1]=labeled
// ---------------------------------------------------------------------------
__global__ __launch_bounds__(32) void crf_reduce_kernel(
    const float* __restrict__ ws_unlab,
    const float* __restrict__ ws_lab,
    float*       __restrict__ out)
{
  if (threadIdx.x == 0) {
    double u = 0.0, l = 0.0;
    for (int b = 0; b < B_SZ; ++b) { u += ws_unlab[b]; l += ws_lab[b]; }
    out[0] = (float)u;
    out[1] = (float)l;
  }
}

extern "C" void kernel_launch(void* const* d_in, const int* in_sizes, int n_in,
                              void* d_out, int out_size, void* d_ws, size_t ws_size,
                              hipStream_t stream) {
  (void)in_sizes; (void)n_in; (void)out_size; (void)ws_size;
  const float* lstm = (const float*)d_in[0];   // (64,512,32) f32
  const int*   lens = (const int*)d_in[1];     // (64,) i32
  const int*   tags = (const int*)d_in[2];     // (64,512) i32
  // d_in[3] = mask: derivable from lens, ignored
  const float* T    = (const float*)d_in[4];   // (32,32) f32
  float* out = (float*)d_out;

  float* ws_unlab = (float*)d_ws;        // 64 floats
  float* ws_lab   = ws_unlab + B_SZ;     // 64 floats

  crf_scan_kernel<<<B_SZ / 16, 32, 0, stream>>>(lstm, lens, T, ws_unlab);
  crf_labeled_kernel<<<B_SZ, 256, 0, stream>>>(lstm, lens, tags, T, ws_lab);
  crf_reduce_kernel<<<1, 32, 0, stream>>>(ws_unlab, ws_lab, out);
}